// NodeEncoder_34746285425415
// MI455X (gfx1250) — compile-verified
//
#include <hip/hip_runtime.h>

typedef float v2f __attribute__((ext_vector_type(2)));
typedef float v8f __attribute__((ext_vector_type(8)));

#define NN_ 100000
#define EE_ 800000
#define DD_ 64
#define LL_ 5
#define GG_ 256
#define BN_EPS_ 1e-5f

// ---------------------------------------------------------------- utilities
__global__ void zero_f32(float* __restrict__ p, long long n) {
    long long i = (long long)blockIdx.x * blockDim.x + threadIdx.x;
    if (i < n) p[i] = 0.0f;
}

// h[i][:] = atom_emb[x[i]][:]   (one thread per float4)
__global__ void embed_nodes(const int* __restrict__ x,
                            const float* __restrict__ atom_emb,
                            float* __restrict__ h, int n) {
    int idx = blockIdx.x * blockDim.x + threadIdx.x;   // n*16 float4 chunks
    if (idx >= n * 16) return;
    int i = idx >> 4, q = idx & 15;
    int a = x[i];
    float4 v = ((const float4*)(atom_emb + (size_t)a * DD_))[q];
    ((float4*)(h + (size_t)i * DD_))[q] = v;
}

// msg = relu(h[src] + bond_emb[eattr]); aggr[dst] += msg  (float4 granularity)
__global__ void edge_msg(const int* __restrict__ src, const int* __restrict__ dst,
                         const int* __restrict__ eattr,
                         const float* __restrict__ bond_emb,
                         const float* __restrict__ h,
                         float* __restrict__ aggr, int e) {
    long long idx = (long long)blockIdx.x * blockDim.x + threadIdx.x; // e*16
    if (idx >= (long long)e * 16) return;
    int ed = (int)(idx >> 4), q = (int)(idx & 15);
    int s = src[ed], d = dst[ed], b = eattr[ed];
    float4 hv = ((const float4*)(h + (size_t)s * DD_))[q];
    float4 bv = ((const float4*)(bond_emb + (size_t)b * DD_))[q];
    float4 m;
    m.x = fmaxf(hv.x + bv.x, 0.0f);
    m.y = fmaxf(hv.y + bv.y, 0.0f);
    m.z = fmaxf(hv.z + bv.z, 0.0f);
    m.w = fmaxf(hv.w + bv.w, 0.0f);
    float* o = aggr + (size_t)d * DD_ + q * 4;
    atomicAdd(o + 0, m.x); atomicAdd(o + 1, m.y);
    atomicAdd(o + 2, m.z); atomicAdd(o + 3, m.w);
}

// ------------------------------------------------------ GEMM1: (h+aggr)@W1+b1
// A: N x 64 (fp32), B: 64 x 128, out t: N x 128.  One wave per 16-row tile.
__global__ __launch_bounds__(128) void gemm1_wmma(
        const float* __restrict__ h, const float* __restrict__ aggr,
        const float* __restrict__ W1, const float* __restrict__ b1,
        float* __restrict__ t, int n) {
    __shared__ float sW[64 * 128];                     // 32 KB weight tile
    for (int i = threadIdx.x; i < (64 * 128) / 4; i += blockDim.x)
        ((float4*)sW)[i] = ((const float4*)W1)[i];
    __syncthreads();

    int wave = threadIdx.x >> 5;
    int lane = threadIdx.x & 31;
    int tile = blockIdx.x * 4 + wave;
    if (tile * 16 >= n) return;                        // wave-uniform exit

    int m = lane & 15;
    int khalf = (lane >> 4) << 1;                      // 0 or 2
    int row = tile * 16 + m;

    // A fragments: lane caches its half of the K=64 slab (32 regs)
    float areg[32];
    const float* hrow = h + (size_t)row * 64;
    const float* grow = aggr + (size_t)row * 64;
#pragma unroll
    for (int ks = 0; ks < 16; ++ks) {
        int k = ks * 4 + khalf;
        areg[2 * ks + 0] = hrow[k + 0] + grow[k + 0];
        areg[2 * ks + 1] = hrow[k + 1] + grow[k + 1];
    }

#pragma unroll
    for (int ct = 0; ct < 8; ++ct) {                   // 8 column tiles of 16
        int nn = ct * 16 + m;
        float bias = b1[nn];
        v8f c;
#pragma unroll
        for (int r = 0; r < 8; ++r) c[r] = bias;
#pragma unroll
        for (int ks = 0; ks < 16; ++ks) {
            int k = ks * 4 + khalf;
            v2f a = {areg[2 * ks], areg[2 * ks + 1]};
            v2f b = {sW[(k + 0) * 128 + nn], sW[(k + 1) * 128 + nn]};
            c = __builtin_amdgcn_wmma_f32_16x16x4_f32(
                    false, a, false, b, (short)0, c, false, false);
        }
        int rbase = tile * 16 + ((lane >> 4) << 3);
#pragma unroll
        for (int r = 0; r < 8; ++r)
            t[(size_t)(rbase + r) * 128 + nn] = c[r];
    }
}

// ---------------------------------------- GEMM2: relu(bn(t))@W2+b2 -> u (Nx64)
__global__ __launch_bounds__(128) void gemm2_wmma(
        const float* __restrict__ t,
        const float* __restrict__ scale1, const float* __restrict__ shift1,
        const float* __restrict__ W2, const float* __restrict__ b2,
        float* __restrict__ u, int n) {
    __shared__ float sW[128 * 64];                     // 32 KB
    for (int i = threadIdx.x; i < (128 * 64) / 4; i += blockDim.x)
        ((float4*)sW)[i] = ((const float4*)W2)[i];
    __syncthreads();

    int wave = threadIdx.x >> 5;
    int lane = threadIdx.x & 31;
    int tile = blockIdx.x * 4 + wave;
    if (tile * 16 >= n) return;

    int m = lane & 15;
    int khalf = (lane >> 4) << 1;
    int row = tile * 16 + m;

    float areg[64];                                    // K=128 slab per lane
    const float* trow = t + (size_t)row * 128;
#pragma unroll
    for (int ks = 0; ks < 32; ++ks) {
        int k = ks * 4 + khalf;
        float z0 = trow[k + 0] * scale1[k + 0] + shift1[k + 0];
        float z1 = trow[k + 1] * scale1[k + 1] + shift1[k + 1];
        areg[2 * ks + 0] = fmaxf(z0, 0.0f);
        areg[2 * ks + 1] = fmaxf(z1, 0.0f);
    }

#pragma unroll
    for (int ct = 0; ct < 4; ++ct) {
        int nn = ct * 16 + m;
        float bias = b2[nn];
        v8f c;
#pragma unroll
        for (int r = 0; r < 8; ++r) c[r] = bias;
#pragma unroll
        for (int ks = 0; ks < 32; ++ks) {
            int k = ks * 4 + khalf;
            v2f a = {areg[2 * ks], areg[2 * ks + 1]};
            v2f b = {sW[(k + 0) * 64 + nn], sW[(k + 1) * 64 + nn]};
            c = __builtin_amdgcn_wmma_f32_16x16x4_f32(
                    false, a, false, b, (short)0, c, false, false);
        }
        int rbase = tile * 16 + ((lane >> 4) << 3);
#pragma unroll
        for (int r = 0; r < 8; ++r)
            u[(size_t)(rbase + r) * 64 + nn] = c[r];
    }
}

// ---------------------------------------------------------------- BN pieces
template <int C>
__global__ void col_stats(const float* __restrict__ data, int n,
                          float* __restrict__ sum, float* __restrict__ sumsq) {
    int j = threadIdx.x;                               // blockDim.x == C
    float s = 0.0f, sq = 0.0f;
    for (int i = blockIdx.x; i < n; i += gridDim.x) {
        float v = data[(size_t)i * C + j];
        s += v; sq += v * v;
    }
    atomicAdd(&sum[j], s);
    atomicAdd(&sumsq[j], sq);
}

__global__ void bn_finalize(const float* __restrict__ sum,
                            const float* __restrict__ sumsq,
                            const float* __restrict__ g,
                            const float* __restrict__ be,
                            float* __restrict__ scale, float* __restrict__ shift,
                            int C, float invN) {
    int j = blockIdx.x * blockDim.x + threadIdx.x;
    if (j < C) {
        float mu = sum[j] * invN;
        float var = sumsq[j] * invN - mu * mu;
        float inv = rsqrtf(var + BN_EPS_);
        float sc = g[j] * inv;
        scale[j] = sc;
        shift[j] = be[j] - mu * sc;
    }
}

__global__ void bn_apply(const float* __restrict__ u,
                         const float* __restrict__ scale,
                         const float* __restrict__ shift,
                         float* __restrict__ h, int n, int do_relu) {
    int idx = blockIdx.x * blockDim.x + threadIdx.x;   // n*64
    if (idx >= n * 64) return;
    int j = idx & 63;
    float v = u[idx] * scale[j] + shift[j];
    if (do_relu) v = fmaxf(v, 0.0f);
    h[idx] = v;
}

// ------------------------------------------- pool: xpool[batch[i]] += h[i]
__global__ void pool_and_out(const float* __restrict__ h,
                             const int* __restrict__ batch,
                             float* __restrict__ out_pool,
                             float* __restrict__ out_h, int n) {
    int idx = blockIdx.x * blockDim.x + threadIdx.x;   // n*16 float4 chunks
    if (idx >= n * 16) return;
    int i = idx >> 4, q = idx & 15;
    float4 v = ((const float4*)(h + (size_t)i * DD_))[q];
    ((float4*)(out_h + (size_t)i * DD_))[q] = v;
    int g = batch[i];
    float* p = out_pool + (size_t)g * DD_ + q * 4;
    atomicAdd(p + 0, v.x); atomicAdd(p + 1, v.y);
    atomicAdd(p + 2, v.z); atomicAdd(p + 3, v.w);
}

static inline int cdiv(long long a, int b) { return (int)((a + b - 1) / b); }

extern "C" void kernel_launch(void* const* d_in, const int* in_sizes, int n_in,
                              void* d_out, int out_size, void* d_ws, size_t ws_size,
                              hipStream_t stream) {
    const int* batch    = (const int*)d_in[0];
    const int* x        = (const int*)d_in[1];
    const int* eidx     = (const int*)d_in[2];
    const int* eattr    = (const int*)d_in[3];
    const float* aemb   = (const float*)d_in[4];
    const float* bemb   = (const float*)d_in[5];
    const float* W1     = (const float*)d_in[6];
    const float* b1     = (const float*)d_in[7];
    const float* g1     = (const float*)d_in[8];
    const float* be1    = (const float*)d_in[9];
    const float* W2     = (const float*)d_in[10];
    const float* b2     = (const float*)d_in[11];
    const float* gbn    = (const float*)d_in[12];
    const float* bbn    = (const float*)d_in[13];

    const int* src = eidx;            // edge_index[0]
    const int* dst = eidx + EE_;      // edge_index[1]

    float* out_pool = (float*)d_out;              // G*64
    float* out_h    = out_pool + GG_ * DD_;       // N*64

    float* ws   = (float*)d_ws;
    float* h    = ws;                               // N*64
    float* aggr = h    + (size_t)NN_ * 64;          // N*64
    float* t    = aggr + (size_t)NN_ * 64;          // N*128
    float* u    = t    + (size_t)NN_ * 128;         // N*64
    float* sum1 = u + (size_t)NN_ * 64;             // 128
    float* sq1  = sum1 + 128;
    float* sc1  = sq1  + 128;
    float* sh1  = sc1  + 128;
    float* sum2 = sh1  + 128;                       // 64
    float* sq2  = sum2 + 64;
    float* sc2  = sq2  + 64;
    float* sh2  = sc2  + 64;

    const int TB = 256;
    const float invN = 1.0f / (float)NN_;
    const int row_tiles = NN_ / 16;                 // 6250 exactly
    const dim3 gemm_grid(cdiv(row_tiles, 4)), gemm_blk(128);

    // fresh accumulators every call (harness replays without re-poisoning)
    zero_f32<<<cdiv(GG_ * DD_, TB), TB, 0, stream>>>(out_pool, GG_ * DD_);
    embed_nodes<<<cdiv((long long)NN_ * 16, TB), TB, 0, stream>>>(x, aemb, h, NN_);

    for (int l = 0; l < LL_; ++l) {
        zero_f32<<<cdiv((long long)NN_ * 64, TB), TB, 0, stream>>>(aggr, (long long)NN_ * 64);
        edge_msg<<<cdiv((long long)EE_ * 16, TB), TB, 0, stream>>>(
            src, dst, eattr, bemb, h, aggr, EE_);

        gemm1_wmma<<<gemm_grid, gemm_blk, 0, stream>>>(
            h, aggr, W1 + (size_t)l * 64 * 128, b1 + (size_t)l * 128, t, NN_);

        zero_f32<<<1, 256, 0, stream>>>(sum1, 256);  // sum1 + sq1
        col_stats<128><<<1024, 128, 0, stream>>>(t, NN_, sum1, sq1);
        bn_finalize<<<1, 128, 0, stream>>>(sum1, sq1, g1 + (size_t)l * 128,
                                           be1 + (size_t)l * 128, sc1, sh1, 128, invN);

        gemm2_wmma<<<gemm_grid, gemm_blk, 0, stream>>>(
            t, sc1, sh1, W2 + (size_t)l * 128 * 64, b2 + (size_t)l * 64, u, NN_);

        zero_f32<<<1, 128, 0, stream>>>(sum2, 128);  // sum2 + sq2
        col_stats<64><<<1024, 64, 0, stream>>>(u, NN_, sum2, sq2);
        bn_finalize<<<1, 64, 0, stream>>>(sum2, sq2, gbn + (size_t)l * 64,
                                          bbn + (size_t)l * 64, sc2, sh2, 64, invN);

        bn_apply<<<cdiv((long long)NN_ * 64, TB), TB, 0, stream>>>(
            u, sc2, sh2, h, NN_, (l < LL_ - 1) ? 1 : 0);
    }

    pool_and_out<<<cdiv((long long)NN_ * 16, TB), TB, 0, stream>>>(
        h, batch, out_pool, out_h, NN_);
}